// ColorConstancy_17274358464665
// MI455X (gfx1250) — compile-verified
//
#include <hip/hip_runtime.h>
#include <cmath>

typedef __attribute__((ext_vector_type(2))) float v2f;
typedef __attribute__((ext_vector_type(8))) float v8f;

#define EPSV 1e-8f
#define HH 512
#define WW 512
#define NPLANE 48                 // 16 batch * 3 channels
#define PLANE_ELEMS (HH*WW)       // 262144
#define TOTAL (NPLANE*PLANE_ELEMS)

#define K0S 13
#define K1S 25
#define K2S 49
#define C0S 6
#define C1S 12
#define C2S 24

#define PAD    24                 // vertical zero-pad rows (max center offset)
#define PROWS  (HH + 2*PAD)       // 560 padded rows per plane
#define PPLANE (PROWS*WW)         // 286720 floats per padded plane

// workspace layout (float offsets)
#define TMP0_OFF  0
#define TMP1_OFF  ((size_t)NPLANE*PPLANE)
#define TMP2_OFF  (2*(size_t)NPLANE*PPLANE)
#define KH_OFF    (3*(size_t)NPLANE*PPLANE)   // hblur coeffs: s0@+0, s1@+64, s2@+128 (unscaled)
#define KV_OFF    (KH_OFF + 256)              // vblur coeffs: zero-padded, w-scaled; s0@+16, s1@+80, s2@+144
#define STATS_OFF (KV_OFF + 256)              // 2 floats per plane: sum, sumsq

// ---------------------------------------------------------------------------
// Prep: zero stats + coefficient regions; extract normalized 1-D Gaussian
// factors from the 2-D depthwise kernels (center row / row-sum).  Writes an
// unscaled copy (horizontal pass) and a zero-padded, weight-scaled copy
// (vertical WMMA pass, so band lookups need no bounds checks or multiplies).
// ---------------------------------------------------------------------------
__global__ void prep_kernel(const float* __restrict__ k0,
                            const float* __restrict__ k1,
                            const float* __restrict__ k2,
                            float* __restrict__ ws) {
  const int t = threadIdx.x;                 // 256 threads
  ws[KH_OFF + t] = 0.0f;
  ws[KV_OFF + t] = 0.0f;
  if (t < 2 * NPLANE) ws[STATS_OFF + t] = 0.0f;
  __syncthreads();                           // zero-fill visible before fills

  if (t == 0) {
    const float* src = k0 + C0S * K0S;       // channel 0, center row
    float s = 0.f;
    for (int i = 0; i < K0S; ++i) s += src[i];
    const float inv = 1.0f / s, wv = (1.00f / 2.25f) * inv;
    for (int i = 0; i < K0S; ++i) {
      ws[KH_OFF +  0 + i] = src[i] * inv;
      ws[KV_OFF + 16 + i] = src[i] * wv;
    }
  } else if (t == 1) {
    const float* src = k1 + C1S * K1S;
    float s = 0.f;
    for (int i = 0; i < K1S; ++i) s += src[i];
    const float inv = 1.0f / s, wv = (0.75f / 2.25f) * inv;
    for (int i = 0; i < K1S; ++i) {
      ws[KH_OFF + 64 + i] = src[i] * inv;
      ws[KV_OFF + 80 + i] = src[i] * wv;
    }
  } else if (t == 2) {
    const float* src = k2 + C2S * K2S;
    float s = 0.f;
    for (int i = 0; i < K2S; ++i) s += src[i];
    const float inv = 1.0f / s, wv = (0.50f / 2.25f) * inv;
    for (int i = 0; i < K2S; ++i) {
      ws[KH_OFF + 128 + i] = src[i] * inv;
      ws[KV_OFF + 144 + i] = src[i] * wv;
    }
  }
}

// ---------------------------------------------------------------------------
// Fused horizontal pass into zero-padded planes.  Grid (2, 560, 48):
// blockIdx.y indexes padded rows; rows outside [0,512) just store zeros so
// the vertical pass never needs bounds checks.
// ---------------------------------------------------------------------------
__global__ __launch_bounds__(256) void hblur_kernel(const float* __restrict__ rgb,
                                                    float* __restrict__ ws) {
  const int tid  = threadIdx.x;
  const int seg  = blockIdx.x;               // 0,1: 256-px column segment
  const int rowp = blockIdx.y;               // 0..559 padded row
  const int p    = blockIdx.z;               // plane
  const int row  = rowp - PAD;

  const size_t o = ((size_t)p * PROWS + rowp) * WW + seg * 256 + tid;
  if (row < 0 || row >= HH) {                // pad row: zeros
    ws[TMP0_OFF + o] = 0.0f;
    ws[TMP1_OFF + o] = 0.0f;
    ws[TMP2_OFF + o] = 0.0f;
    return;
  }

  __shared__ float sh[256 + 2 * C2S];        // 304 halo'd log values
  __shared__ float shk[K0S + K1S + K2S];

  const float* krn = ws + KH_OFF;
  if (tid < K0S)                  shk[tid] = krn[tid];
  else if (tid < K0S + K1S)       shk[tid] = krn[64 + (tid - K0S)];
  else if (tid < K0S + K1S + K2S) shk[tid] = krn[128 + (tid - K0S - K1S)];

  const int xbase = seg * 256 - C2S;
  const float* src = rgb + (size_t)p * PLANE_ELEMS + (size_t)row * WW;
  for (int i = tid; i < 256 + 2 * C2S; i += 256) {
    const int x = xbase + i;
    float v = 0.0f;
    if (x >= 0 && x < WW) v = logf(src[x] + EPSV);   // zero-padded conv input
    sh[i] = v;
  }
  __syncthreads();

  float a0 = 0.f, a1 = 0.f, a2 = 0.f;
  const int b0 = tid + C2S - C0S;
  #pragma unroll
  for (int t = 0; t < K0S; ++t) a0 = fmaf(shk[t], sh[b0 + t], a0);
  const int b1 = tid + C2S - C1S;
  #pragma unroll
  for (int t = 0; t < K1S; ++t) a1 = fmaf(shk[K0S + t], sh[b1 + t], a1);
  #pragma unroll
  for (int t = 0; t < K2S; ++t) a2 = fmaf(shk[K0S + K1S + t], sh[tid + t], a2);

  ws[TMP0_OFF + o] = a0;
  ws[TMP1_OFF + o] = a1;
  ws[TMP2_OFF + o] = a2;
}

// ---------------------------------------------------------------------------
// Vertical pass via V_WMMA_F32_16X16X4_F32, one wave32 per 16x16 tile.
// D(16x16) = sum_j A_band(16x4) x In(4x16); A[m][j] = w_s*k1d_s[j-m].
// A 16x4 layout: lane<16 -> M=lane, v0:K=j0+0, v1:K=j0+1;
//                lane>=16 -> M=lane-16, v0:K=j0+2, v1:K=j0+3.
// B 4x16 layout: v0 = rows {j0+0 | j0+2}, v1 = rows {j0+1 | j0+3}.
// All loads are branch-free: planes are zero-padded in rows, coefficients are
// zero-padded in LDS (shkp already offset by base+16-m).
// ---------------------------------------------------------------------------
template <int K>
__device__ __forceinline__ v8f accum_scale(v8f acc,
                                           const float* __restrict__ tpx,
                                           const float* shkp,
                                           int ko0, int ko1) {
  #pragma unroll
  for (int j0 = 0; j0 < K + 15; j0 += 4) {   // K+15 is a multiple of 4
    v2f a, b;
    a.x = shkp[j0 + ko0];
    a.y = shkp[j0 + ko1];
    b.x = tpx[(j0 + ko0) * WW];
    b.y = tpx[(j0 + ko1) * WW];
    acc = __builtin_amdgcn_wmma_f32_16x16x4_f32(false, a, false, b,
                                                (short)0, acc, false, false);
  }
  return acc;
}

__global__ __launch_bounds__(256) void vblur_wmma_kernel(const float* __restrict__ rgb,
                                                         float* __restrict__ ws,
                                                         float* __restrict__ refl_out) {
  __shared__ float shk[256];                 // padded, w-scaled coefficients
  shk[threadIdx.x] = ws[KV_OFF + threadIdx.x];
  __syncthreads();

  const int lane = threadIdx.x & 31;
  const int wave = threadIdx.x >> 5;
  const int tile = blockIdx.x * 8 + wave;    // 48*32*32 tiles exactly
  const int p  = tile >> 10;
  const int ty = (tile >> 5) & 31;
  const int tx = tile & 31;
  const int y0 = ty * 16, x0 = tx * 16;

  const int m   = lane & 15;
  const int hi  = lane >> 4;
  const int ko0 = hi * 2;
  const int ko1 = ko0 + 1;
  const int x   = x0 + m;

  const size_t pb = (size_t)p * PPLANE;

  v8f acc = {};
  acc = accum_scale<K0S>(acc,
      ws + TMP0_OFF + pb + (size_t)(y0 - C0S + PAD) * WW + x, &shk[ 16 - m],       ko0, ko1);
  acc = accum_scale<K1S>(acc,
      ws + TMP1_OFF + pb + (size_t)(y0 - C1S + PAD) * WW + x, &shk[ 80 - m],       ko0, ko1);
  acc = accum_scale<K2S>(acc,
      ws + TMP2_OFF + pb + (size_t)(y0 - C2S + PAD) * WW + x, &shk[144 - m],       ko0, ko1);

  // refl = log(rgb+eps) - illum ; write refl into d_out; reduce sum / sumsq.
  const float* src = rgb + (size_t)p * PLANE_ELEMS;
  float* ro = refl_out + (size_t)p * PLANE_ELEMS;
  float lsum = 0.0f, lsq = 0.0f;
  #pragma unroll
  for (int r = 0; r < 8; ++r) {              // C/D: v_r = rows {r | r+8}
    const int y = y0 + r + hi * 8;
    const float lg = logf(src[(size_t)y * WW + x] + EPSV);
    const float rf = lg - acc[r];
    ro[(size_t)y * WW + x] = rf;
    lsum += rf;
    lsq  += rf * rf;
  }
  #pragma unroll
  for (int o = 16; o > 0; o >>= 1) {
    lsum += __shfl_xor(lsum, o, 32);
    lsq  += __shfl_xor(lsq,  o, 32);
  }
  if (lane == 0) {
    atomicAdd(ws + STATS_OFF + 2 * p,     lsum);
    atomicAdd(ws + STATS_OFF + 2 * p + 1, lsq);
  }
}

// ---------------------------------------------------------------------------
// Normalize in place: (refl-mean)/(std+eps) -> exp -> clip.  std is ddof=1.
// ---------------------------------------------------------------------------
__global__ __launch_bounds__(256) void norm_kernel(float* __restrict__ out,
                                                   const float* __restrict__ ws) {
  const size_t i4 = (size_t)blockIdx.x * 256 + threadIdx.x;   // float4 index
  const int p = (int)((i4 * 4) / PLANE_ELEMS);

  const float sum = ws[STATS_OFF + 2 * p];
  const float sq  = ws[STATS_OFF + 2 * p + 1];
  const float Nf  = (float)PLANE_ELEMS;
  const float mean = sum / Nf;
  const float var  = (sq - sum * sum / Nf) / (Nf - 1.0f);
  const float inv  = 1.0f / (sqrtf(fmaxf(var, 0.0f)) + EPSV);

  float4 v = reinterpret_cast<float4*>(out)[i4];
  v.x = fminf(fmaxf(expf((v.x - mean) * inv), 0.0f), 1.0f);
  v.y = fminf(fmaxf(expf((v.y - mean) * inv), 0.0f), 1.0f);
  v.z = fminf(fmaxf(expf((v.z - mean) * inv), 0.0f), 1.0f);
  v.w = fminf(fmaxf(expf((v.w - mean) * inv), 0.0f), 1.0f);
  reinterpret_cast<float4*>(out)[i4] = v;
}

extern "C" void kernel_launch(void* const* d_in, const int* in_sizes, int n_in,
                              void* d_out, int out_size, void* d_ws, size_t ws_size,
                              hipStream_t stream) {
  const float* rgb = (const float*)d_in[0];
  const float* k0  = (const float*)d_in[1];
  const float* k1  = (const float*)d_in[2];
  const float* k2  = (const float*)d_in[3];
  float* ws  = (float*)d_ws;
  float* out = (float*)d_out;

  prep_kernel<<<1, 256, 0, stream>>>(k0, k1, k2, ws);
  hblur_kernel<<<dim3(2, PROWS, NPLANE), 256, 0, stream>>>(rgb, ws);
  vblur_wmma_kernel<<<(NPLANE * 32 * 32) / 8, 256, 0, stream>>>(rgb, ws, out);
  norm_kernel<<<TOTAL / 4 / 256, 256, 0, stream>>>(out, ws);
}